// All_old_33998961115770
// MI455X (gfx1250) — compile-verified
//
#include <hip/hip_runtime.h>

// ---------------------------------------------------------------------------
// Gather + segment-sum (sorted segment ids), HBM-bandwidth bound.
//   out[c, :] = sum_{m : segment_ids[m]==c} raw_h[member_idx[m], :]
//
// Pass 1 (boundary kernel): one streaming sweep over segment_ids builds
//   starts[c] = lower_bound(segment_ids, c) for c in [0, n_cliques], each
//   entry written exactly once (deterministic, no atomics).
// Pass 2: one wave32 per clique; 32 lanes x float4 = one 512B row per issue.
//   Rows are gathered with CDNA5 async global->LDS copies (ASYNCcnt),
//   accumulated from LDS, and stored with non-temporal hints so L2 stays
//   dedicated to the 256MB gather table (avg 3x reuse per row).
// ---------------------------------------------------------------------------

#define WAVES_PER_BLOCK 8
#define DEPTH 8   // async rows in flight per wave (8 * 512B * 8 waves = 32KB LDS)
#define HIDDEN 128

typedef float f4 __attribute__((ext_vector_type(4)));
typedef int   v4i __attribute__((vector_size(16)));   // matches builtin's pointee type

#if defined(__has_builtin)
#  if __has_builtin(__builtin_amdgcn_global_load_async_to_lds_b128)
#    define HAVE_ASYNC_LDS 1
#  endif
#  if __has_builtin(__builtin_amdgcn_s_wait_asynccnt)
#    define HAVE_WAIT_ASYNC 1
#  endif
#endif

#if defined(HAVE_ASYNC_LDS)
typedef __attribute__((address_space(1))) v4i* g_v4i_ptr;  // global src
typedef __attribute__((address_space(3))) v4i* l_v4i_ptr;  // LDS dst
#endif

// ---- Pass 1: segment boundary sweep ---------------------------------------
__global__ __launch_bounds__(256)
void seg_starts_kernel(const int* __restrict__ segment_ids,
                       int*       __restrict__ starts,   // n_cliques + 1 entries
                       int n_members, int n_cliques)
{
    const int i = blockIdx.x * blockDim.x + threadIdx.x;
    if (i >= n_members) return;
    const int s_i = segment_ids[i];
    if (i == 0) {
        for (int c = 0; c <= s_i; ++c) starts[c] = 0;
    } else {
        const int s_p = segment_ids[i - 1];
        if (s_p != s_i)
            for (int c = s_p + 1; c <= s_i; ++c) starts[c] = i;  // covers empties
    }
    if (i == n_members - 1) {
        for (int c = s_i + 1; c <= n_cliques; ++c) starts[c] = n_members;
    }
}

// ---- Pass 2: gather + segment sum -----------------------------------------
__global__ __launch_bounds__(WAVES_PER_BLOCK * 32)
void clique_segsum_kernel(const float* __restrict__ raw_h,
                          const int*   __restrict__ member_idx,
                          const int*   __restrict__ segment_ids,
                          const int*   __restrict__ starts,   // may be nullptr
                          float*       __restrict__ out,
                          int n_members, int n_cliques)
{
#if defined(HAVE_ASYNC_LDS)
    __shared__ f4 smem[WAVES_PER_BLOCK][DEPTH][32];
#endif
    const int wave = threadIdx.x >> 5;
    const int lane = threadIdx.x & 31;
    const int c    = blockIdx.x * WAVES_PER_BLOCK + wave;
    if (c >= n_cliques) return;

    int beg, end;
    if (starts != nullptr) {
        beg = starts[c];
        end = starts[c + 1];
    } else {
        // Fallback: uniform binary searches (no scratch available).
        int lo = 0, hi = n_members;
        while (lo < hi) {
            int mid = (lo + hi) >> 1;
            if (segment_ids[mid] < c) lo = mid + 1; else hi = mid;
        }
        beg = lo;
        int e_lo = beg, e_hi = n_members;
        while (e_lo < e_hi) {
            int mid = (e_lo + e_hi) >> 1;
            if (segment_ids[mid] < c + 1) e_lo = mid + 1; else e_hi = mid;
        }
        end = e_lo;
    }

    f4 acc = (f4){0.f, 0.f, 0.f, 0.f};

#if defined(HAVE_ASYNC_LDS)
    // Chunked async gather: up to DEPTH rows in flight, then drain from LDS.
    for (int base = beg; base < end; base += DEPTH) {
        const int cnt = (end - base) < DEPTH ? (end - base) : DEPTH;
        // Lane-parallel fetch of this chunk's indices (one load), then shuffle.
        int myidx = 0;
        if (lane < cnt) myidx = member_idx[base + lane];
        for (int s = 0; s < cnt; ++s) {
            const int idx = __shfl(myidx, s, 32);
            const float* src = raw_h + (size_t)idx * HIDDEN + lane * 4;
            __builtin_amdgcn_global_load_async_to_lds_b128(
                (g_v4i_ptr)(void*)const_cast<float*>(src),
                (l_v4i_ptr)(void*)&smem[wave][s][lane],
                0, 0);
        }
#if defined(HAVE_WAIT_ASYNC)
        __builtin_amdgcn_s_wait_asynccnt(0);
#else
        asm volatile("s_wait_asynccnt 0x0" ::: "memory");
#endif
        asm volatile("" ::: "memory");  // compiler barrier before LDS reads
        for (int s = 0; s < cnt; ++s) {
            f4 r = smem[wave][s][lane];
            acc += r;
        }
    }
#else
    // Fallback: direct vectorized gather (global_load_b128 per lane).
    for (int m = beg; m < end; ++m) {
        const int idx = member_idx[m];
        const f4* row = (const f4*)(raw_h + (size_t)idx * HIDDEN);
        acc += row[lane];
    }
#endif

    // Streaming (non-temporal) store: keep L2 for the gather table.
    f4* dst = (f4*)(out + (size_t)c * HIDDEN) + lane;
    __builtin_nontemporal_store(acc, dst);
}

extern "C" void kernel_launch(void* const* d_in, const int* in_sizes, int n_in,
                              void* d_out, int out_size, void* d_ws, size_t ws_size,
                              hipStream_t stream) {
    (void)n_in;
    const float* raw_h       = (const float*)d_in[0];
    const int*   member_idx  = (const int*)d_in[1];
    const int*   segment_ids = (const int*)d_in[2];
    float*       out         = (float*)d_out;

    const int n_members = in_sizes[1];
    const int n_cliques = out_size / HIDDEN;

    int* starts = nullptr;
    const size_t need = (size_t)(n_cliques + 1) * sizeof(int);
    if (d_ws != nullptr && ws_size >= need) {
        starts = (int*)d_ws;
        const int b1 = (n_members + 255) / 256;
        seg_starts_kernel<<<b1, 256, 0, stream>>>(segment_ids, starts,
                                                  n_members, n_cliques);
    }

    const int blocks = (n_cliques + WAVES_PER_BLOCK - 1) / WAVES_PER_BLOCK;
    clique_segsum_kernel<<<blocks, WAVES_PER_BLOCK * 32, 0, stream>>>(
        raw_h, member_idx, segment_ids, starts, out, n_members, n_cliques);
}